// PhysicsLoss_7009386627594
// MI455X (gfx1250) — compile-verified
//
#include <hip/hip_runtime.h>
#include <hip/hip_bf16.h>
#include <stdint.h>

// Problem: B=256, N=50000, E=128000.
// ws layout: [acc: B*N f32 (51.2MB)] [head_partials: double] [mass_partials: double]

#define TILE 256
#define BCHUNK 32
#define MASS_BLOCKS 2048
#define RED_THREADS 256

// ---------------- CDNA5 async global->LDS path (guarded) ----------------
#ifdef __has_builtin
#  if __has_builtin(__builtin_amdgcn_global_load_async_to_lds_b32)
#    define HAVE_ASYNC_LDS 1
#  endif
#  if __has_builtin(__builtin_amdgcn_s_wait_asynccnt)
#    define WAIT_ASYNC(n) __builtin_amdgcn_s_wait_asynccnt(n)
#  endif
#endif
#ifndef HAVE_ASYNC_LDS
#  define HAVE_ASYNC_LDS 0
#endif
#ifndef WAIT_ASYNC
#  if HAVE_ASYNC_LDS
#    define WAIT_ASYNC(n) asm volatile("s_wait_asynccnt %0" ::"i"(n) : "memory")
#  else
#    define WAIT_ASYNC(n)
#  endif
#endif

#if HAVE_ASYNC_LDS
// Per hipcc diagnostic: first param is '__device__ int *' (addrspace(1) int*),
// so the b32 builtin takes typed int pointers, not void*.
typedef __attribute__((address_space(1))) int gas_int;
typedef __attribute__((address_space(3))) int las_int;
__device__ __forceinline__ void async_g2l_b32(const void* gptr, void* lptr) {
  // per-lane: copies 4 bytes from global address gptr to LDS address lptr; tracked by ASYNCcnt
  __builtin_amdgcn_global_load_async_to_lds_b32(
      (gas_int*)(uintptr_t)gptr,
      (las_int*)(unsigned int)(uintptr_t)lptr, /*offset=*/0, /*cpol=*/0);
}
#endif

// ---------------- Kernel 1: acc = -(true_d_norm*d_scale + d_mean) ----------------
__global__ __launch_bounds__(256) void init_acc_kernel(
    const float* __restrict__ td, const float* __restrict__ d_scale,
    const float* __restrict__ d_mean, float* __restrict__ acc, int N) {
  int n = blockIdx.x * blockDim.x + threadIdx.x;
  int b = blockIdx.y;
  if (n < N) {
    size_t i = (size_t)b * N + n;
    acc[i] = -__builtin_fmaf(td[i], d_scale[n], d_mean[n]);
  }
}

// ---------------- Kernel 2: edge pass (scatter-add + head loss) ----------------
__global__ __launch_bounds__(TILE) void edge_kernel(
    const float* __restrict__ pf, const float* __restrict__ pp,
    const float* __restrict__ f_mean, const float* __restrict__ f_scale,
    const float* __restrict__ p_mean, const float* __restrict__ p_scale,
    const float* __restrict__ elev, const float* __restrict__ R,
    const int* __restrict__ src, const int* __restrict__ dst,
    float* __restrict__ acc, double* __restrict__ head_partials,
    int E, int N, int B) {
  const int tid = threadIdx.x;
  const int e = blockIdx.x * TILE + tid;
  const bool valid = (e < E);
  const int b0 = blockIdx.y * BCHUNK;
  int nb = B - b0;
  if (nb > BCHUNK) nb = BCHUNK;
  if (nb < 0) nb = 0;

  // Per-edge invariants live in registers across the whole b-loop.
  int ss = 0, dd = 0;
  float fm = 0.f, fs = 0.f, r = 0.f;
  float ps_s = 0.f, bias_s = 0.f, ps_d = 0.f, bias_d = 0.f;
  if (valid) {
    ss = src[e];
    dd = dst[e];
    fm = f_mean[e];
    fs = f_scale[e];
    r = R[e];
    ps_s = p_scale[ss];
    bias_s = p_mean[ss] + elev[ss];  // H = P + elev; fold mean+elev per node
    ps_d = p_scale[dd];
    bias_d = p_mean[dd] + elev[dd];
  }

  double sum_head = 0.0;

#if HAVE_ASYNC_LDS
  __shared__ float s_q[2][TILE];
  if (valid && nb > 0) async_g2l_b32(pf + (size_t)b0 * E + e, &s_q[0][tid]);
  for (int bi = 0; bi < nb; ++bi) {
    const int b = b0 + bi;
    if (bi + 1 < nb) {
      if (valid) async_g2l_b32(pf + (size_t)(b + 1) * E + e, &s_q[(bi + 1) & 1][tid]);
      WAIT_ASYNC(1);  // current buffer complete (in-order), next in flight
    } else {
      WAIT_ASYNC(0);
    }
    if (valid) {
      float q = __builtin_fmaf(s_q[bi & 1][tid], fs, fm);
#else
  for (int bi = 0; bi < nb; ++bi) {
    const int b = b0 + bi;
    if (valid) {
      if (bi + 1 < nb) __builtin_prefetch(pf + (size_t)(b + 1) * E + e, 0, 0);
      float q = __builtin_fmaf(pf[(size_t)b * E + e], fs, fm);
#endif
      float* accb = acc + (size_t)b * N;
      atomicAdd(accb + dd, q);   // q_in at dst
      atomicAdd(accb + ss, -q);  // -q_out at src
      // gathers hit a 200KB L2-resident row of pred_p_norm for this b
      float hs = __builtin_fmaf(pp[(size_t)b * N + ss], ps_s, bias_s);
      float hd = __builtin_fmaf(pp[(size_t)b * N + dd], ps_d, bias_d);
      float aq = fabsf(q);
      float fric = r * q * exp2f(0.852f * log2f(aq));  // |0|^0.852 -> exp2(-inf)=0
      float t = (hs - hd) - fric;
      sum_head += (double)t * (double)t;
    }
  }

  __shared__ double red[TILE];
  red[tid] = sum_head;
  __syncthreads();
  for (int off = TILE / 2; off > 0; off >>= 1) {
    if (tid < off) red[tid] += red[tid + off];
    __syncthreads();
  }
  if (tid == 0) head_partials[(size_t)blockIdx.y * gridDim.x + blockIdx.x] = red[0];
}

// ---------------- Kernel 3: loss_mass partial sums (acc holds q_net - d) ----------------
__global__ __launch_bounds__(RED_THREADS) void mass_kernel(
    const float* __restrict__ acc, double* __restrict__ partials, long long total) {
  double s = 0.0;
  const long long stride = (long long)gridDim.x * blockDim.x;
  const long long i0 = (long long)blockIdx.x * blockDim.x + threadIdx.x;
  const long long total4 = total >> 2;
  const float4* acc4 = (const float4*)acc;
  for (long long i = i0; i < total4; i += stride) {
    float4 v = acc4[i];
    s += (double)v.x * v.x + (double)v.y * v.y + (double)v.z * v.z + (double)v.w * v.w;
  }
  for (long long i = (total4 << 2) + i0; i < total; i += stride) {
    float v = acc[i];
    s += (double)v * v;
  }
  __shared__ double red[RED_THREADS];
  red[threadIdx.x] = s;
  __syncthreads();
  for (int off = RED_THREADS / 2; off > 0; off >>= 1) {
    if (threadIdx.x < off) red[threadIdx.x] += red[threadIdx.x + off];
    __syncthreads();
  }
  if (threadIdx.x == 0) partials[blockIdx.x] = red[0];
}

// ---------------- Kernel 4: finalize ----------------
__global__ __launch_bounds__(RED_THREADS) void finalize_kernel(
    const double* __restrict__ headp, int nh, const double* __restrict__ massp, int nm,
    float* __restrict__ out, double inv_mass, double inv_head) {
  double sh = 0.0, sm = 0.0;
  for (int i = threadIdx.x; i < nh; i += blockDim.x) sh += headp[i];
  for (int i = threadIdx.x; i < nm; i += blockDim.x) sm += massp[i];
  __shared__ double r1[RED_THREADS], r2[RED_THREADS];
  r1[threadIdx.x] = sh;
  r2[threadIdx.x] = sm;
  __syncthreads();
  for (int off = RED_THREADS / 2; off > 0; off >>= 1) {
    if (threadIdx.x < off) {
      r1[threadIdx.x] += r1[threadIdx.x + off];
      r2[threadIdx.x] += r2[threadIdx.x + off];
    }
    __syncthreads();
  }
  if (threadIdx.x == 0) {
    out[0] = (float)(r2[0] * inv_mass);  // loss_mass
    out[1] = (float)(r1[0] * inv_head);  // loss_head
  }
}

extern "C" void kernel_launch(void* const* d_in, const int* in_sizes, int n_in,
                              void* d_out, int out_size, void* d_ws, size_t ws_size,
                              hipStream_t stream) {
  const float* pred_p  = (const float*)d_in[0];
  const float* pred_f  = (const float*)d_in[1];
  const float* true_d  = (const float*)d_in[2];
  const float* p_mean  = (const float*)d_in[3];
  const float* p_scale = (const float*)d_in[4];
  const float* f_mean  = (const float*)d_in[5];
  const float* f_scale = (const float*)d_in[6];
  const float* d_mean  = (const float*)d_in[7];
  const float* d_scale = (const float*)d_in[8];
  const float* elev    = (const float*)d_in[9];
  const float* R       = (const float*)d_in[10];
  const int*   edge    = (const int*)d_in[11];

  const int N = in_sizes[3];
  const int E = in_sizes[5];
  const int B = in_sizes[0] / N;
  const int* src = edge;      // edge_index row 0
  const int* dst = edge + E;  // edge_index row 1

  float* acc = (float*)d_ws;
  size_t accBytes = (((size_t)B * N * sizeof(float)) + 15) & ~(size_t)15;
  const int ex = (E + TILE - 1) / TILE;
  const int by = (B + BCHUNK - 1) / BCHUNK;
  const int nh = ex * by;
  double* headp = (double*)((char*)d_ws + accBytes);
  double* massp = headp + nh;

  dim3 gInit((unsigned)((N + 255) / 256), (unsigned)B);
  init_acc_kernel<<<gInit, 256, 0, stream>>>(true_d, d_scale, d_mean, acc, N);

  dim3 gEdge((unsigned)ex, (unsigned)by);
  edge_kernel<<<gEdge, TILE, 0, stream>>>(pred_f, pred_p, f_mean, f_scale, p_mean, p_scale,
                                          elev, R, src, dst, acc, headp, E, N, B);

  long long total = (long long)B * N;
  mass_kernel<<<MASS_BLOCKS, RED_THREADS, 0, stream>>>(acc, massp, total);

  finalize_kernel<<<1, RED_THREADS, 0, stream>>>(headp, nh, massp, MASS_BLOCKS, (float*)d_out,
                                                 1.0 / (double)total,
                                                 1.0 / ((double)B * (double)E));
}